// Proto_Layer_27393301414373
// MI455X (gfx1250) — compile-verified
//
#include <hip/hip_runtime.h>

typedef float v2f __attribute__((ext_vector_type(2)));
typedef float v8f __attribute__((ext_vector_type(8)));

#define T_STEPS   8
#define WAY       64
#define SHOT      4
#define NQ        1024      // queries per t (64 way * 16 query)
#define C         512
#define QPB       64        // queries per block
#define CK        64        // K chunk staged in LDS
#define PITCH     68        // floats per LDS row (bank-conflict-free for b64 frag loads)

__global__ __launch_bounds__(256, 4)
void proto_dist_kernel(const float* __restrict__ q,
                       const float* __restrict__ s,
                       float* __restrict__ out)
{
    __shared__ float ldsQ[QPB * PITCH];   // 17408 B
    __shared__ float ldsP[WAY * PITCH];   // 17408 B
    __shared__ float qnp[QPB][4];
    __shared__ float pnp[WAY][4];
    __shared__ float qn[QPB];
    __shared__ float pn[WAY];

    const int tid = threadIdx.x;
    const int t   = blockIdx.x >> 4;      // 8 time-steps
    const int qb  = blockIdx.x & 15;      // 16 query blocks of 64

    const float* Qg = q + ((size_t)t * NQ + (size_t)qb * QPB) * C;
    const float* Sg = s + (size_t)t * (WAY * SHOT) * C;
    float*       Og = out + ((size_t)t * NQ + (size_t)qb * QPB) * WAY;

    // staging assignment: 4 threads per row, each covers 4 float4s per chunk
    const int row = tid >> 2;             // 0..63 (query row AND way row)
    const int sub = tid & 3;

    // wave tiling: 8 waves; wm selects 16 query rows, wn selects 32 way cols
    const int lane = tid & 31;
    const int w    = tid >> 5;
    const int wm   = w & 3;
    const int n0   = (w >> 2) * 32;
    const int half = lane >> 4;
    const int l15  = lane & 15;

    v8f acc0 = {};
    v8f acc1 = {};
    float qn_priv = 0.0f;
    float pn_priv = 0.0f;

    for (int ck = 0; ck < C; ck += CK) {
        __syncthreads();   // previous chunk's compute finished before overwrite

        // ---- stage Q chunk (64 x 64) and accumulate query sq-norm ----
        #pragma unroll
        for (int i = 0; i < 4; ++i) {
            const int col = (sub + 4 * i) * 4;
            const float4 v = *(const float4*)(Qg + (size_t)row * C + ck + col);
            *(float4*)&ldsQ[row * PITCH + col] = v;
            qn_priv += v.x * v.x + v.y * v.y + v.z * v.z + v.w * v.w;
        }

        // ---- stage P chunk = mean over 4 shots, accumulate proto sq-norm ----
        #pragma unroll
        for (int i = 0; i < 4; ++i) {
            const int col = (sub + 4 * i) * 4;
            const float* sp = Sg + (size_t)(row * SHOT) * C + ck + col;
            const float4 a = *(const float4*)(sp);
            const float4 b = *(const float4*)(sp + C);
            const float4 c = *(const float4*)(sp + 2 * C);
            const float4 d = *(const float4*)(sp + 3 * C);
            float4 m;
            m.x = 0.25f * (a.x + b.x + c.x + d.x);
            m.y = 0.25f * (a.y + b.y + c.y + d.y);
            m.z = 0.25f * (a.z + b.z + c.z + d.z);
            m.w = 0.25f * (a.w + b.w + c.w + d.w);
            *(float4*)&ldsP[row * PITCH + col] = m;
            pn_priv += m.x * m.x + m.y * m.y + m.z * m.z + m.w * m.w;
        }

        __syncthreads();   // chunk staged

        // ---- WMMA over the chunk: each wave does a 16x32 tile ----
        #pragma unroll
        for (int kk = 0; kk < CK; kk += 4) {
            const int k = kk + 2 * half;   // lanes 0-15: K={0,1}; lanes 16-31: K={2,3}
            const v2f a  = *(const v2f*)&ldsQ[(wm * 16 + l15) * PITCH + k];
            const v2f b0 = *(const v2f*)&ldsP[(n0 + l15) * PITCH + k];
            const v2f b1 = *(const v2f*)&ldsP[(n0 + 16 + l15) * PITCH + k];
            acc0 = __builtin_amdgcn_wmma_f32_16x16x4_f32(
                false, a, false, b0, (short)0, acc0, false, false);
            acc1 = __builtin_amdgcn_wmma_f32_16x16x4_f32(
                false, a, false, b1, (short)0, acc1, false, false);
        }
    }

    // ---- reduce the 4 partial norms per row ----
    qnp[row][sub] = qn_priv;
    pnp[row][sub] = pn_priv;
    __syncthreads();
    if (tid < 64) {
        qn[tid] = qnp[tid][0] + qnp[tid][1] + qnp[tid][2] + qnp[tid][3];
    } else if (tid < 128) {
        const int r = tid - 64;
        pn[r] = pnp[r][0] + pnp[r][1] + pnp[r][2] + pnp[r][3];
    }
    __syncthreads();

    // ---- combine: dist = 2*QP - ||q||^2 - ||p||^2, write out ----
    const float pn0 = pn[n0 + l15];
    const float pn1 = pn[n0 + 16 + l15];
    #pragma unroll
    for (int i = 0; i < 8; ++i) {
        const int r   = wm * 16 + i + 8 * half;   // C/D layout: M = i + 8*(lane>=16)
        const float qr = qn[r];
        Og[(size_t)r * WAY + n0 + l15]      = 2.0f * acc0[i] - qr - pn0;
        Og[(size_t)r * WAY + n0 + 16 + l15] = 2.0f * acc1[i] - qr - pn1;
    }
}

extern "C" void kernel_launch(void* const* d_in, const int* in_sizes, int n_in,
                              void* d_out, int out_size, void* d_ws, size_t ws_size,
                              hipStream_t stream)
{
    const float* qf = (const float*)d_in[0];   // (8, 1024, 512) f32
    const float* sf = (const float*)d_in[1];   // (8, 256, 512)  f32
    float* out = (float*)d_out;                // (8, 1024, 64)  f32

    const int t = in_sizes[0] / (NQ * C);      // = 8
    dim3 grid(t * (NQ / QPB));                 // 128 blocks
    dim3 block(256);                           // 8 waves (wave32)
    hipLaunchKernelGGL(proto_dist_kernel, grid, block, 0, stream, qf, sf, out);
}